// ModelNew_80908593923311
// MI455X (gfx1250) — compile-verified
//
#include <hip/hip_runtime.h>
#include <cmath>

typedef __attribute__((ext_vector_type(2))) float v2f;
typedef __attribute__((ext_vector_type(8))) float v8f;

#define N_IMG 32
#define C_IN  64
#define H     128
#define W     128
#define C_OUT 64
#define OH    126
#define OW    126
#define CKK   576            // C_IN * 9, GEMM K

// LDS input tile, channel-last: sx[pos][c], pos = row*TCOLS+col
//   4 output rows + 2 halo, 32 output cols + 2 halo
#define TROWS 6
#define TCOLS 34
#define NPOS  (TROWS * TCOLS)        // 204 positions
#define CPAD  68                     // channel stride (pad 64->68: conflict-free b64)
#define STAGE_CNT (C_IN * NPOS)      // 13056 elements staged
#define LDS_CNT   (NPOS * CPAD)      // 13872 floats = 55.5 KB

// ---------------------------------------------------------------------------
// Weight pre-pack: wpack[kk][nt][lane] = { w[n][k'], w[n][k'+1] }  (v2f)
//   kk = 0..143 (K-step of 4 in permuted-K order k' = r*64 + c)
//   n = nt*16 + (lane&15),  k' = 4*kk + 2*(lane>>4) + e
//   orig k = c*9 + r  with c = k' & 63, r = k' >> 6
// ---------------------------------------------------------------------------
__global__ __launch_bounds__(256) void pack_weights(
    const float* __restrict__ w, float* __restrict__ wpack)
{
    int idx  = blockIdx.x * 256 + threadIdx.x;     // 0 .. 36863
    int e    = idx & 1;
    int lane = (idx >> 1) & 31;
    int nt   = (idx >> 6) & 3;
    int kk   = idx >> 8;
    int n    = nt * 16 + (lane & 15);
    int kp   = 4 * kk + 2 * (lane >> 4) + e;
    int c    = kp & 63;
    int r    = kp >> 6;
    wpack[idx] = w[n * CKK + c * 9 + r];
}

// ---------------------------------------------------------------------------
// Main fused kernel: implicit-GEMM conv (f32 WMMA) + bias + min(ch) + tanh^2
// ---------------------------------------------------------------------------
__global__ __launch_bounds__(128) void conv_min_tanh_wmma(
    const float* __restrict__ x,
    const float* __restrict__ wpack,
    const float* __restrict__ bias,
    float* __restrict__ out)
{
    __shared__ float sx[LDS_CNT];

    const int tid  = threadIdx.x;
    const int t    = blockIdx.x;
    const int tw   = t & 3;           // 4 column tiles of 32 (128 >= 126)
    const int trow = (t >> 2) & 31;   // 32 row groups of 4 (128 >= 126)
    const int img  = t >> 7;          // 32 images
    const int ow0  = tw * 32;
    const int oh0  = trow * 4;

    // ---- cooperative stage: coalesced global reads, channel-last LDS writes ----
    const float* ximg = x + (size_t)img * (C_IN * H * W);
    for (int i = tid; i < STAGE_CNT; i += 128) {   // i order: [c][row][col]
        int c   = i / NPOS;
        int pos = i - c * NPOS;                    // row*TCOLS + col
        int row = pos / TCOLS;
        int col = pos - row * TCOLS;
        int ih  = oh0 + row; if (ih > H - 1) ih = H - 1;   // halo clamp (masked later)
        int iw  = ow0 + col; if (iw > W - 1) iw = W - 1;
        sx[pos * CPAD + c] = ximg[(c * H + ih) * W + iw];
    }
    __syncthreads();

    const int wave  = tid >> 5;       // 4 waves: one output row each
    const int lane  = tid & 31;
    const int mlane = lane & 15;      // M within tile / N within tile
    const int khalf = lane >> 4;      // K-half: lanes 16-31 take K+2,K+3
    const int oh    = oh0 + wave;

    if (oh < OH) {                    // wave-uniform -> EXEC all-ones for WMMA
        v8f acc0[4] = {{}, {}, {}, {}};   // M tile 0 (cols ow0+0..15), nt = 0..3
        v8f acc1[4] = {{}, {}, {}, {}};   // M tile 1 (cols ow0+16..31)

        const v2f* wp = (const v2f*)wpack + lane;   // fixed base, immediate offsets

        for (int r0 = 0; r0 < 9; ++r0) {            // 3x3 tap index (uniform)
            int kh = (r0 * 11) >> 5;                // r0 / 3
            int kw = r0 - kh * 3;
            // fixed base per r0: all 32 A-loads are immediate-offset ds_load_b64
            const float* ab = sx + ((wave + kh) * TCOLS + mlane + kw) * CPAD + 2 * khalf;
            const v2f*   wr = wp + r0 * 2048;       // r0*16*4*32 v2f

            #pragma unroll
            for (int cc = 0; cc < 16; ++cc) {       // channels c = 4*cc + 2*khalf + e
                v2f a0 = *(const v2f*)(ab + cc * 16);             // {k', k'+1}
                v2f a1 = *(const v2f*)(ab + cc * 16 + 16 * CPAD); // col+16 tile

                #pragma unroll
                for (int nt = 0; nt < 4; ++nt) {
                    v2f b = wr[cc * 128 + nt * 32];
                    acc0[nt] = __builtin_amdgcn_wmma_f32_16x16x4_f32(
                        false, a0, false, b, (short)0, acc0[nt], false, false);
                    acc1[nt] = __builtin_amdgcn_wmma_f32_16x16x4_f32(
                        false, a1, false, b, (short)0, acc1[nt], false, false);
                }
            }
        }

        // ---- bias add + min over the 4 channel tiles ----
        float bb[4];
        #pragma unroll
        for (int nt = 0; nt < 4; ++nt) bb[nt] = bias[nt * 16 + mlane];

        v8f mn0, mn1;
        #pragma unroll
        for (int j = 0; j < 8; ++j) {
            mn0[j] = fminf(fminf(acc0[0][j] + bb[0], acc0[1][j] + bb[1]),
                           fminf(acc0[2][j] + bb[2], acc0[3][j] + bb[3]));
            mn1[j] = fminf(fminf(acc1[0][j] + bb[0], acc1[1][j] + bb[1]),
                           fminf(acc1[2][j] + bb[2], acc1[3][j] + bb[3]));
        }

        // ---- min across 16 lanes of each half (C/D layout: halves hold M, M+8) ----
        #pragma unroll
        for (int off = 1; off <= 8; off <<= 1) {
            #pragma unroll
            for (int j = 0; j < 8; ++j) {
                mn0[j] = fminf(mn0[j], __shfl_xor(mn0[j], off, 32));
                mn1[j] = fminf(mn1[j], __shfl_xor(mn1[j], off, 32));
            }
        }

        // ---- tanh(tanh(min)) + masked store; lane0 -> M 0..7, lane16 -> M 8..15 ----
        if (mlane == 0) {
            float* orow = out + ((size_t)img * OH + oh) * OW;
            #pragma unroll
            for (int j = 0; j < 8; ++j) {
                int p0 = ow0 + khalf * 8 + j;
                int p1 = p0 + 16;
                if (p0 < OW) orow[p0] = tanhf(tanhf(mn0[j]));
                if (p1 < OW) orow[p1] = tanhf(tanhf(mn1[j]));
            }
        }
    }
}

extern "C" void kernel_launch(void* const* d_in, const int* in_sizes, int n_in,
                              void* d_out, int out_size, void* d_ws, size_t ws_size,
                              hipStream_t stream) {
    const float* x    = (const float*)d_in[0];
    const float* wgt  = (const float*)d_in[1];
    const float* bias = (const float*)d_in[2];
    float* out        = (float*)d_out;
    float* wpack      = (float*)d_ws;   // needs C_OUT*CKK*4 = 147456 bytes

    // 1) pack weights into WMMA B-fragment order (L2-resident, 36864 elems)
    hipLaunchKernelGGL(pack_weights, dim3((C_OUT * CKK) / 256), dim3(256), 0, stream,
                       wgt, wpack);

    // 2) fused conv + min + tanh^2
    hipLaunchKernelGGL(conv_min_tanh_wmma, dim3(N_IMG * 32 * 4), dim3(128), 0, stream,
                       x, wpack, bias, out);
}